// GSGPP_13683765805698
// MI455X (gfx1250) — compile-verified
//
#include <hip/hip_runtime.h>
#include <hip/hip_bf16.h>

typedef __attribute__((ext_vector_type(2))) float v2f;
typedef __attribute__((ext_vector_type(8))) float v8f;
typedef int v4i __attribute__((vector_size(16)));          // matches builtin proto
typedef __attribute__((address_space(1))) v4i AS1_v4i;     // global
typedef __attribute__((address_space(3))) v4i AS3_v4i;     // LDS

#if __has_builtin(__builtin_amdgcn_global_load_async_to_lds_b128)
#define HAVE_ASYNC_LDS 1
#else
#define HAVE_ASYNC_LDS 0
#endif

// ---------------------------------------------------------------------------
// Utility kernels
// ---------------------------------------------------------------------------
__global__ void zero_f32(float* __restrict__ p, long n) {
  long i = (long)blockIdx.x * blockDim.x + threadIdx.x;
  if (i < n) p[i] = 0.0f;
}

// cnt[idx[i]] += 1  (degree per node / node count per graph)
__global__ void count_kernel(const int* __restrict__ idx, float* __restrict__ cnt, int n) {
  int i = blockIdx.x * blockDim.x + threadIdx.x;
  if (i < n) atomicAdd(&cnt[idx[i]], 1.0f);
}

// agg[dst[e]] += h[src[e]]   (per edge, 4 features per thread, coalesced)
__global__ void gather_kernel(const int* __restrict__ src, const int* __restrict__ dst,
                              const float* __restrict__ h, int ld_in,
                              float* __restrict__ agg, int ld_out, int E, int d) {
  int chunks = d >> 2;
  long idx = (long)blockIdx.x * blockDim.x + threadIdx.x;
  if (idx >= (long)E * chunks) return;
  int e = (int)(idx / chunks);
  int c = (int)(idx % chunks) << 2;
  int s = src[e], t = dst[e];
  const float4 v = *(const float4*)(h + (size_t)s * ld_in + c);
  float* o = agg + (size_t)t * ld_out + c;
  atomicAdd(o + 0, v.x);
  atomicAdd(o + 1, v.y);
  atomicAdd(o + 2, v.z);
  atomicAdd(o + 3, v.w);
}

// pooled[batch[i]] += h[i]
__global__ void pool_kernel(const float* __restrict__ h, const int* __restrict__ batch,
                            float* __restrict__ pooled, int n, int d) {
  int chunks = d >> 2;
  long idx = (long)blockIdx.x * blockDim.x + threadIdx.x;
  if (idx >= (long)n * chunks) return;
  int i = (int)(idx / chunks);
  int c = (int)(idx % chunks) << 2;
  int g = batch[i];
  const float4 v = *(const float4*)(h + (size_t)i * d + c);
  float* o = pooled + (size_t)g * d + c;
  atomicAdd(o + 0, v.x);
  atomicAdd(o + 1, v.y);
  atomicAdd(o + 2, v.z);
  atomicAdd(o + 3, v.w);
}

// ---------------------------------------------------------------------------
// Fused SAGE GEMM:
//   out[16 rows x d_out] = relu( (agg/deg) @ Wl + bias + root @ Wr )
// One block = 16 rows; each wave owns two 16x16 col tiles (32 cols).
// blockDim.x == d_out (256 -> 8 waves, 64 -> 2 waves); d_in, d_out powers of 2.
// Weight matrices are staged 16 K-rows at a time into LDS via
// GLOBAL_LOAD_ASYNC_TO_LDS_B128 (ASYNCcnt) when available, so the WMMA
// B-fragments come from conflict-free ds_load_b32 instead of scalar vmem.
// root == nullptr => skip Wr term (used for the final FC on pooled means).
// Safe to alias out with agg: a block reads only its own 16 rows into LDS
// before writing those same rows.
// ---------------------------------------------------------------------------
__launch_bounds__(256)
__global__ void sage_gemm_wmma(const float* __restrict__ root, int ld_root,
                               const float* __restrict__ agg, int ld_agg,
                               const float* __restrict__ denom,
                               const float* __restrict__ Wl,
                               const float* __restrict__ Wr,
                               const float* __restrict__ bias,
                               float* __restrict__ out,
                               int nrows, int d_in, int d_out, int do_relu) {
  __shared__ float sX[16][260];   // root rows (+4 pad: conflict-free b64 A reads)
  __shared__ float sM[16][260];   // mean rows
  __shared__ float sW[16][264];   // weight chunk (+8 pad: conflict-free b32 B reads)
  __shared__ float sInv[16];

  const int row0 = blockIdx.x * 16;
  const int tid = threadIdx.x;

  if (tid < 16) {
    int r = row0 + tid;
    float dg = (r < nrows) ? denom[r] : 1.0f;
    sInv[tid] = 1.0f / fmaxf(dg, 1.0f);
  }
  __syncthreads();

  // ---- stage this block's 16 rows of root / mean features (shift-only math) ----
  {
    const int dshift = __builtin_ctz((unsigned)d_in);
    const int dmask = d_in - 1;
    for (int idx = tid; idx < 16 * d_in; idx += d_out) {
      int r = idx >> dshift;
      int k = idx & dmask;
      int grow = row0 + r;
      float xv = 0.0f, mv = 0.0f;
      if (grow < nrows) {
        if (root) xv = root[(size_t)grow * ld_root + k];
        mv = agg[(size_t)grow * ld_agg + k] * sInv[r];
      }
      sX[r][k] = xv;
      sM[r][k] = mv;
    }
  }

  const int lane = tid & 31;
  const int wave = tid >> 5;
  const int col0 = wave * 32;           // this wave's 32 output columns
  const int halfk = lane >> 4;          // 0: K=k0,k0+1  1: K=k0+2,k0+3  (ISA A/B layout)
  const int mrow = lane & 15;

  // Per-thread weight-staging coordinates, computed once (cvec = d_out/4 is pow2;
  // d_out threads x 4 passes cover the 16 x d_out chunk, 16B per transfer).
  const int cvec = d_out >> 2;
  const int cshift = __builtin_ctz((unsigned)cvec);
  const int sr = tid >> cshift;               // starting row 0..3
  const int sc = (tid & (cvec - 1)) << 2;     // float column

  v8f c0 = {};
  v8f c1 = {};

  for (int phase = (root ? 0 : 1); phase < 2; ++phase) {
    const float* sA = (phase == 0) ? &sX[0][0] : &sM[0][0];
    const float* W  = (phase == 0) ? Wr : Wl;

    for (int kc = 0; kc < d_in; kc += 16) {
      __syncthreads();                 // staged rows (first iter) / sW consumers done
      // ---- stage 16 x d_out weight chunk into LDS (no div in hot loop) ----
#pragma unroll
      for (int pass = 0; pass < 4; ++pass) {
        int r = sr + (pass << 2);
        const float* g = W + (size_t)(kc + r) * d_out + sc;
#if HAVE_ASYNC_LDS
        AS1_v4i* gp = (AS1_v4i*)(unsigned long long)(uintptr_t)g;
        AS3_v4i* lp = (AS3_v4i*)(unsigned)(unsigned long long)(uintptr_t)&sW[r][sc];
        __builtin_amdgcn_global_load_async_to_lds_b128(gp, lp, 0, 0);
#else
        *(float4*)&sW[r][sc] = *(const float4*)g;
#endif
      }
#if HAVE_ASYNC_LDS
#if __has_builtin(__builtin_amdgcn_s_wait_asynccnt)
      __builtin_amdgcn_s_wait_asynccnt(0);
#else
      asm volatile("s_wait_asynccnt 0x0" ::: "memory");
#endif
#endif
      __syncthreads();

      // ---- 4 WMMA K-steps over the staged chunk ----
#pragma unroll
      for (int k0 = 0; k0 < 16; k0 += 4) {
        int ka = k0 + 2 * halfk;
        v2f a = *(const v2f*)&sA[mrow * 260 + kc + ka];   // A: 16x4 fragment (ds b64)
        v2f b0, b1;                                        // B: 4x16 fragments (ds b32)
        b0[0] = sW[ka][col0 + mrow];
        b0[1] = sW[ka + 1][col0 + mrow];
        b1[0] = sW[ka][col0 + 16 + mrow];
        b1[1] = sW[ka + 1][col0 + 16 + mrow];
        c0 = __builtin_amdgcn_wmma_f32_16x16x4_f32(false, a, false, b0, (short)0, c0, false, false);
        c1 = __builtin_amdgcn_wmma_f32_16x16x4_f32(false, a, false, b1, (short)0, c1, false, false);
      }
    }
  }

  const float bv0 = bias[col0 + mrow];
  const float bv1 = bias[col0 + 16 + mrow];
#pragma unroll
  for (int j = 0; j < 8; ++j) {
    int r = j + 8 * halfk;               // C/D layout: vgpr j -> M=j (lanes 0-15) / j+8 (16-31)
    int grow = row0 + r;
    if (grow < nrows) {
      float v0 = c0[j] + bv0;
      float v1 = c1[j] + bv1;
      if (do_relu) { v0 = fmaxf(v0, 0.0f); v1 = fmaxf(v1, 0.0f); }
      out[(size_t)grow * d_out + col0 + mrow] = v0;
      out[(size_t)grow * d_out + col0 + 16 + mrow] = v1;
    }
  }
}

// ---------------------------------------------------------------------------
static inline void launch_zero(float* p, long n, hipStream_t s) {
  zero_f32<<<(unsigned)((n + 255) / 256), 256, 0, s>>>(p, n);
}

extern "C" void kernel_launch(void* const* d_in, const int* in_sizes, int n_in,
                              void* d_out, int out_size, void* d_ws, size_t ws_size,
                              hipStream_t stream) {
  const int N = 50000, E = 800000, G = 512, F_IN = 128, H = 256, OUT = 64;

  const float* x     = (const float*)d_in[0];
  const int*   ei    = (const int*)d_in[1];
  const int*   batch = (const int*)d_in[2];
  const float* Wl1 = (const float*)d_in[3];
  const float* bl1 = (const float*)d_in[4];
  const float* Wr1 = (const float*)d_in[5];
  const float* Wl2 = (const float*)d_in[6];
  const float* bl2 = (const float*)d_in[7];
  const float* Wr2 = (const float*)d_in[8];
  const float* Wl3 = (const float*)d_in[9];
  const float* bl3 = (const float*)d_in[10];
  const float* Wr3 = (const float*)d_in[11];
  const float* fcw = (const float*)d_in[12];
  const float* fcb = (const float*)d_in[13];

  const int* src = ei;       // edge_index[0]
  const int* dst = ei + E;   // edge_index[1]

  // Workspace layout (floats): two NxH ping-pong buffers + deg + pooled + counts
  float* ws = (float*)d_ws;
  size_t off = 0;
  float* bufA   = ws + off; off += (size_t)N * H;
  float* bufB   = ws + off; off += (size_t)N * H;
  float* deg    = ws + off; off += (size_t)N;
  float* pooled = ws + off; off += (size_t)G * H;
  float* counts = ws + off; off += (size_t)G;
  (void)ws_size; (void)in_sizes; (void)n_in; (void)out_size;

  const int mblocks = (N + 15) / 16;   // 3125

  // Degree per node, node-count per graph (fresh every call)
  launch_zero(deg, N, stream);
  launch_zero(counts, G, stream);
  count_kernel<<<(E + 255) / 256, 256, 0, stream>>>(dst, deg, E);
  count_kernel<<<(N + 255) / 256, 256, 0, stream>>>(batch, counts, N);

  // ---- Layer 1: agg(x) -> bufA (stride H), h1 = gemm -> bufA (in place) ----
  launch_zero(bufA, (long)N * H, stream);
  {
    long t = (long)E * (F_IN / 4);
    gather_kernel<<<(unsigned)((t + 255) / 256), 256, 0, stream>>>(src, dst, x, F_IN, bufA, H, E, F_IN);
  }
  sage_gemm_wmma<<<mblocks, 256, 0, stream>>>(x, F_IN, bufA, H, deg, Wl1, Wr1, bl1,
                                              bufA, N, F_IN, H, 1);

  // ---- Layer 2: agg(h1) -> bufB, h2 = gemm -> bufB ----
  launch_zero(bufB, (long)N * H, stream);
  {
    long t = (long)E * (H / 4);
    gather_kernel<<<(unsigned)((t + 255) / 256), 256, 0, stream>>>(src, dst, bufA, H, bufB, H, E, H);
  }
  sage_gemm_wmma<<<mblocks, 256, 0, stream>>>(bufA, H, bufB, H, deg, Wl2, Wr2, bl2,
                                              bufB, N, H, H, 1);

  // ---- Layer 3: agg(h2) -> bufA, h3 = gemm -> bufA (no relu) ----
  launch_zero(bufA, (long)N * H, stream);
  {
    long t = (long)E * (H / 4);
    gather_kernel<<<(unsigned)((t + 255) / 256), 256, 0, stream>>>(src, dst, bufB, H, bufA, H, E, H);
  }
  sage_gemm_wmma<<<mblocks, 256, 0, stream>>>(bufB, H, bufA, H, deg, Wl3, Wr3, bl3,
                                              bufA, N, H, H, 0);

  // ---- Global mean pool + FC ----
  launch_zero(pooled, (long)G * H, stream);
  {
    long t = (long)N * (H / 4);
    pool_kernel<<<(unsigned)((t + 255) / 256), 256, 0, stream>>>(bufA, batch, pooled, N, H);
  }
  // out[G,OUT] = (pooled/counts) @ fc_w + fc_b   (root disabled => Wl path only)
  sage_gemm_wmma<<<(G + 15) / 16, OUT, 0, stream>>>(nullptr, 0, pooled, H, counts,
                                                    fcw, nullptr, fcb,
                                                    (float*)d_out, G, H, OUT, 0);
}